// TopKActivation_50706383896773
// MI455X (gfx1250) — compile-verified
//
#include <hip/hip_runtime.h>
#include <stdint.h>

// TopK sparse activation, one row (16384 fp32) per workgroup.
// - Row staged in LDS via gfx1250 GLOBAL_LOAD_ASYNC_TO_LDS_B128 (ASYNCcnt path),
//   with non-temporal hint (single-use stream, keep 192MB L2 clean).
// - Exact k-th-largest via 4-round radix select on order-preserving uint32 keys;
//   per-wave privatized histograms + wave32 shfl suffix scan (3 barriers/round).
// - Output: coalesced non-temporal b128 stores.
// HBM traffic is the theoretical minimum: 512 MB total -> ~22 us at 23.3 TB/s.

#define ROWLEN 16384
#define NT     512
#define NWAVE  (NT / 32)                 // wave32 -> 16 waves per block
#define CHUNKS (ROWLEN / (NT * 4))       // 8 x b128 chunks per thread

typedef float v4f __attribute__((ext_vector_type(4)));

__device__ __forceinline__ uint32_t fkey(uint32_t b) {
  // Order-preserving map fp32 -> uint32 (ascending float => ascending key).
  uint32_t m = (uint32_t)((int32_t)b >> 31) | 0x80000000u;
  return b ^ m;
}

__global__ __launch_bounds__(NT) void topk_row_kernel(
    const float* __restrict__ x, const int* __restrict__ kp,
    float* __restrict__ out) {
  __shared__ __align__(16) uint32_t s_keys[ROWLEN];   // 64 KB: raw fp32 bits of the row
  __shared__ uint32_t s_histw[NWAVE][256];            // 16 KB: per-wave privatized histograms
  __shared__ uint32_t s_selbin, s_selrem, s_tie;

  const int tid = threadIdx.x;
  const int wid = tid >> 5;                           // wave32 index
  const uint64_t rowoff = (uint64_t)blockIdx.x * (uint64_t)ROWLEN;
  const float* src = x + rowoff;

  // ---- Stage 1: async DMA of the whole 64 KB row into LDS (non-temporal) ----
  const uint32_t lds_base = (uint32_t)(uintptr_t)&s_keys[0];
#pragma unroll
  for (int j = 0; j < CHUNKS; ++j) {
    uint32_t boff = (uint32_t)(tid + j * NT) * 16u;   // byte offset of this b128 chunk
    uint32_t ldst = lds_base + boff;                  // per-lane LDS dest address
    asm volatile("global_load_async_to_lds_b128 %0, %1, %2 th:TH_LOAD_NT"
                 :
                 : "v"(ldst), "v"(boff), "s"(src)
                 : "memory");
  }
  asm volatile("s_wait_asynccnt 0" ::: "memory");
  __syncthreads();

  const int k = *kp;                                  // uniform scalar load
  const uint4* kv = (const uint4*)s_keys;
  v4f* dst = (v4f*)(out + rowoff);

  if (k <= 0 || k >= ROWLEN) {
    // Reference degenerates to plain relu(x) in both cases.
#pragma unroll
    for (int j = 0; j < CHUNKS; ++j) {
      uint4 v = kv[tid + j * NT];
      v4f o;
      o.x = fmaxf(__uint_as_float(v.x), 0.0f);
      o.y = fmaxf(__uint_as_float(v.y), 0.0f);
      o.z = fmaxf(__uint_as_float(v.z), 0.0f);
      o.w = fmaxf(__uint_as_float(v.w), 0.0f);
      __builtin_nontemporal_store(o, &dst[tid + j * NT]);
    }
    return;
  }

  // ---- Stage 2: 4-round radix select (8 bits/round) for the exact k-th largest key ----
  uint32_t prefix = 0u, pmask = 0u, remaining = (uint32_t)k;
  for (int round = 0; round < 4; ++round) {
    const int shift = 24 - 8 * round;

    for (int i = tid; i < NWAVE * 256; i += NT) ((uint32_t*)s_histw)[i] = 0u;
    __syncthreads();

#pragma unroll
    for (int j = 0; j < CHUNKS; ++j) {
      uint4 v = kv[tid + j * NT];
      uint32_t u;
      u = fkey(v.x); if ((u & pmask) == prefix) atomicAdd(&s_histw[wid][(u >> shift) & 255u], 1u);
      u = fkey(v.y); if ((u & pmask) == prefix) atomicAdd(&s_histw[wid][(u >> shift) & 255u], 1u);
      u = fkey(v.z); if ((u & pmask) == prefix) atomicAdd(&s_histw[wid][(u >> shift) & 255u], 1u);
      u = fkey(v.w); if ((u & pmask) == prefix) atomicAdd(&s_histw[wid][(u >> shift) & 255u], 1u);
    }
    __syncthreads();

    // Wave 0 alone: reduce 16 per-wave histograms (8 bins/lane, 32 lanes = 256 bins),
    // in-register suffix sums + wave32 shfl suffix scan, then pick the bin.
    if (wid == 0) {
      const int lane = tid;                           // 0..31
      uint32_t c[8];
#pragma unroll
      for (int i = 0; i < 8; ++i) {
        uint32_t s = 0;
#pragma unroll
        for (int w = 0; w < NWAVE; ++w) s += s_histw[w][lane * 8 + i];
        c[i] = s;
      }
      uint32_t suf[8];                                // suf[i] = c[i] + ... + c[7]
      suf[7] = c[7];
#pragma unroll
      for (int i = 6; i >= 0; --i) suf[i] = suf[i + 1] + c[i];
      const uint32_t tot = suf[0];
      // Inclusive suffix scan of lane totals across the wave (higher lane = higher bins).
      uint32_t acc = tot;
#pragma unroll
      for (int off = 1; off < 32; off <<= 1) {
        uint32_t t = __shfl_down(acc, off, 32);
        if (lane + off < 32) acc += t;
      }
      const uint32_t above = acc - tot;               // count in bins owned by lanes > lane
      // Bin b = lane*8+i: S(b)=count(>=b)=above+suf[i]; count(>b)=S(b)-c[i].
#pragma unroll
      for (int i = 0; i < 8; ++i) {
        uint32_t S  = above + suf[i];
        uint32_t gt = S - c[i];
        if (S >= remaining && gt < remaining) {       // unique transition bin
          s_selbin = (uint32_t)(lane * 8 + i);
          s_selrem = remaining - gt;
        }
      }
    }
    __syncthreads();

    prefix |= s_selbin << shift;
    pmask  |= (255u << shift);
    remaining = s_selrem;
    // s_selbin/s_selrem are not rewritten until after two more barriers next round.
  }
  // prefix == exact key of the k-th largest element; 'remaining' = #ties at the
  // threshold that belong to the kept set.

  if (tid == 0) s_tie = 0u;
  __syncthreads();
  const uint32_t u_thr = prefix;
  const uint32_t ties  = remaining;

  // ---- Stage 3: output pass (coalesced non-temporal b128 stores) ----
  auto pick = [&](uint32_t bits) -> float {
    uint32_t u = fkey(bits);
    if (u > u_thr) return fmaxf(__uint_as_float(bits), 0.0f);
    if (u == u_thr) {
      if (atomicAdd(&s_tie, 1u) < ties) return fmaxf(__uint_as_float(bits), 0.0f);
    }
    return 0.0f;
  };

#pragma unroll
  for (int j = 0; j < CHUNKS; ++j) {
    uint4 v = kv[tid + j * NT];
    v4f o;
    o.x = pick(v.x);
    o.y = pick(v.y);
    o.z = pick(v.z);
    o.w = pick(v.w);
    __builtin_nontemporal_store(o, &dst[tid + j * NT]);
  }
}

extern "C" void kernel_launch(void* const* d_in, const int* in_sizes, int n_in,
                              void* d_out, int out_size, void* d_ws, size_t ws_size,
                              hipStream_t stream) {
  const float* x   = (const float*)d_in[0];
  const int*   kp  = (const int*)d_in[1];
  float*       out = (float*)d_out;
  (void)n_in; (void)out_size; (void)d_ws; (void)ws_size;

  const int rows = in_sizes[0] / ROWLEN;   // 4096 for the reference shapes
  topk_row_kernel<<<dim3(rows), dim3(NT), 0, stream>>>(x, kp, out);
}